// PatchyLayerCNNTopLast_14714557956451
// MI455X (gfx1250) — compile-verified
//
#include <hip/hip_runtime.h>

typedef float v2f __attribute__((ext_vector_type(2)));
typedef float v8f __attribute__((ext_vector_type(8)));

// Problem constants (from the reference):
constexpr int Bn   = 64;    // batches
constexpr int Lseq = 4096;  // sequence length
constexpr int Cn   = 64;    // channels
constexpr int Pn   = 3000;  // patches
constexpr int Kn   = 8;     // taps per patch

// One block per patch p. 128 threads = 4 waves; wave w handles batches
// [16w, 16w+16). Each wave computes out[b0..b0+15, p] with 128 chained
// V_WMMA_F32_16X16X4_F32 ops (fp32-exact), B-operand broadcast across N.
__global__ __launch_bounds__(128) void patchy_wmma_kernel(
    const float* __restrict__ y,        // (B, L, C)
    const int*   __restrict__ patches,  // (P, K)
    const float* __restrict__ wmult,    // (P, K, C)
    const float* __restrict__ wbias,    // (P,)
    float*       __restrict__ out)      // (B, P)
{
    __shared__ float lw[Kn * Cn];  // 512 floats = 2KB: W_MULT[p,:,:]

    const int p = blockIdx.x;
    const int t = threadIdx.x;

    // Stage this patch's weights into LDS (128 threads x 4 floats).
    const float* wp = wmult + (size_t)p * (Kn * Cn);
#pragma unroll
    for (int i = 0; i < (Kn * Cn) / 128; ++i)
        lw[t + i * 128] = wp[t + i * 128];
    __syncthreads();

    const int lane = t & 31;
    const int m    = lane & 15;        // A-matrix row within the 16-b tile
    const int koff = (lane >> 4) * 2;  // K slots {0,1} (lanes 0-15) or {2,3} (lanes 16-31)
    const int b0   = (t >> 5) * 16;    // batch tile base for this wave

    const float* ybase = y + (size_t)(b0 + m) * (Lseq * Cn);

    v8f acc0 = {};
    v8f acc1 = {};

#pragma unroll 1
    for (int k = 0; k < Kn; ++k) {
        const int row = patches[p * Kn + k];         // uniform -> scalar load
        const float* yr = ybase + row * Cn + koff;   // gathered y row (this lane's K slots)
        const float* wr = lw + k * Cn + koff;        // weights from LDS
#pragma unroll
        for (int c0 = 0; c0 < Cn; c0 += 8) {
            v2f a0, b0v, a1, b1v;
            a0.x  = yr[c0];     a0.y  = yr[c0 + 1];   // global_load_b64
            b0v.x = wr[c0];     b0v.y = wr[c0 + 1];   // ds_load_b64
            a1.x  = yr[c0 + 4]; a1.y  = yr[c0 + 5];
            b1v.x = wr[c0 + 4]; b1v.y = wr[c0 + 5];
            // D = A(16x4 f32) x B(4x16 f32, w broadcast over N) + C
            acc0 = __builtin_amdgcn_wmma_f32_16x16x4_f32(
                false, a0, false, b0v, (short)0, acc0, false, false);
            acc1 = __builtin_amdgcn_wmma_f32_16x16x4_f32(
                false, a1, false, b1v, (short)0, acc1, false, false);
        }
    }

    // All 16 D columns are identical; column N=0 lives in lane 0 (M=0..7 in
    // VGPR 0..7) and lane 16 (M=8..15).
    if ((lane & 15) == 0) {
        const float bias = wbias[p];
        const int   mb   = b0 + (lane >> 4) * 8;
#pragma unroll
        for (int r = 0; r < 8; ++r) {
            float v = acc0[r] + acc1[r] + bias;
            v = (v < 0.0f) ? 0.1f * v : v;   // leaky_relu, slope 0.1
            out[(size_t)(mb + r) * Pn + p] = v;
        }
    }
}

extern "C" void kernel_launch(void* const* d_in, const int* in_sizes, int n_in,
                              void* d_out, int out_size, void* d_ws, size_t ws_size,
                              hipStream_t stream) {
    (void)in_sizes; (void)n_in; (void)d_ws; (void)ws_size; (void)out_size;
    const float* y       = (const float*)d_in[0];  // (B, L, C) fp32
    const int*   patches = (const int*)  d_in[1];  // (P, K, 1) int32
    const float* wmult   = (const float*)d_in[2];  // (1, P, K, C) fp32
    const float* wbias   = (const float*)d_in[3];  // (1, P) fp32
    float*       out     = (float*)d_out;          // (B, P) fp32

    dim3 grid(Pn), block(128);
    patchy_wmma_kernel<<<grid, block, 0, stream>>>(y, patches, wmult, wbias, out);
}